// CharEmbedding_69630009802942
// MI455X (gfx1250) — compile-verified
//
#include <hip/hip_runtime.h>

// char-LSTM on gfx1250: 16 words per wave, fp32 WMMA 16x16x4 for both GEMMs.
// Gate matrix processed one 16-column block at a time (acc = 4 tiles = 32 VGPRs).
// Bias table in LDS, loaded AFTER the GEMMs (short live range -> no spills);
// bank-conflict-free pitches (36 / 68); __launch_bounds__(256,4) pins VGPRs <=256.

#define VOCAB   100
#define E       32
#define HDIM    64
#define G4H     256      // 4*H
#define MAXLEN  16
#define WAVES   8        // waves per block -> 128 words per block
#define EPITCH  36       // emb row pitch
#define WIHP    36       // W_ih row pitch  (bank = 36n+2h mod 64: conflict-free)
#define WHHP    68       // W_hh row pitch  (bank = 4n+2h  mod 64: conflict-free)
#define HPITCH  68       // per-wave h buffer pitch (68 = 4*17 -> 16B alignable)

typedef __attribute__((ext_vector_type(2))) float v2f;
typedef __attribute__((ext_vector_type(4))) float v4f;
typedef __attribute__((ext_vector_type(8))) float v8f;

__device__ __forceinline__ float sigm(float x) {
    return 1.0f / (1.0f + __expf(-x));
}

__global__ __launch_bounds__(WAVES * 32, 4)
void char_lstm_wmma(const float* __restrict__ emb,
                    const float* __restrict__ Wih,
                    const float* __restrict__ Whh,
                    const float* __restrict__ bih,
                    const float* __restrict__ bhh,
                    const int*   __restrict__ chars,
                    const int*   __restrict__ lengths,
                    float*       __restrict__ out)
{
    __shared__ float s_emb[VOCAB * EPITCH];        // 14.4 KB
    __shared__ float s_wih[G4H * WIHP];            // 36.9 KB
    __shared__ float s_whh[G4H * WHHP];            // 69.6 KB
    __shared__ float s_bias[G4H];                  //  1.0 KB
    __shared__ float s_h[WAVES][16 * HPITCH];      // 34.8 KB

    const int tid  = threadIdx.x;
    const int lane = tid & 31;
    const int wave = tid >> 5;
    const int half = lane >> 4;     // 0 -> lanes 0-15, 1 -> lanes 16-31
    const int n    = lane & 15;     // tile column (B/C/D) == word row (A)
    const int base = (blockIdx.x * WAVES + wave) * 16;

    // ---- stage tables into LDS (whole block cooperates) ----
    for (int i = tid; i < VOCAB * E; i += WAVES * 32)
        s_emb[(i / E) * EPITCH + (i % E)] = emb[i];
    for (int i = tid; i < G4H * E; i += WAVES * 32)
        s_wih[(i / E) * WIHP + (i % E)] = Wih[i];
    for (int i = tid; i < G4H * HDIM; i += WAVES * 32)
        s_whh[(i / HDIM) * WHHP + (i % HDIM)] = Whh[i];
    for (int i = tid; i < G4H; i += WAVES * 32)
        s_bias[i] = bih[i] + bhh[i];
    for (int i = lane; i < 16 * HPITCH; i += 32)   // h0 = 0
        s_h[wave][i] = 0.0f;

    // ---- per-lane constants ----
    int llen[8];
    #pragma unroll
    for (int r = 0; r < 8; ++r)
        llen[r] = lengths[base + 8 * half + r];

    v8f creg[4];   // cell state, D layout, 4 column blocks of 16
    v8f hreg[4];   // hidden state, D layout (mirrors s_h)
    #pragma unroll
    for (int cb = 0; cb < 4; ++cb)
        #pragma unroll
        for (int r = 0; r < 8; ++r) { creg[cb][r] = 0.0f; hreg[cb][r] = 0.0f; }

    __syncthreads();

    #pragma unroll 1
    for (int t = 0; t < MAXLEN; ++t) {
        const int c    = chars[(base + n) * MAXLEN + t];
        const int xoff = c * EPITCH + 2 * half;    // x A-fragment base (lds idx)

        // one 16-wide column block of H at a time; acc holds i,f,g,o tiles there
        #pragma unroll
        for (int cb = 0; cb < 4; ++cb) {
            v8f acc[4];
            #pragma unroll
            for (int g = 0; g < 4; ++g)
                #pragma unroll
                for (int r = 0; r < 8; ++r) acc[g][r] = 0.0f;

            // gates[:, g*64 + cb*16 + n] += X[16,32] @ Wih^T
            #pragma unroll
            for (int kk = 0; kk < E / 4; ++kk) {
                v2f a = *(const v2f*)&s_emb[xoff + kk * 4];
                #pragma unroll
                for (int g = 0; g < 4; ++g) {
                    v2f b = *(const v2f*)&s_wih[(g * 64 + cb * 16 + n) * WIHP
                                                + kk * 4 + 2 * half];
                    acc[g] = __builtin_amdgcn_wmma_f32_16x16x4_f32(
                        false, a, false, b, (short)0, acc[g], false, false);
                }
            }

            // gates[:, ...] += h[16,64] @ Whh^T  (reads PREVIOUS h from LDS)
            #pragma unroll
            for (int kk = 0; kk < HDIM / 4; ++kk) {
                v2f a = *(const v2f*)&s_h[wave][n * HPITCH + kk * 4 + 2 * half];
                #pragma unroll
                for (int g = 0; g < 4; ++g) {
                    v2f b = *(const v2f*)&s_whh[(g * 64 + cb * 16 + n) * WHHP
                                                + kk * 4 + 2 * half];
                    acc[g] = __builtin_amdgcn_wmma_f32_16x16x4_f32(
                        false, a, false, b, (short)0, acc[g], false, false);
                }
            }

            // bias loaded here (short live range, after the WMMA-pressure region)
            float b0 = s_bias[0 * 64 + cb * 16 + n];
            float b1 = s_bias[1 * 64 + cb * 16 + n];
            float b2 = s_bias[2 * 64 + cb * 16 + n];
            float b3 = s_bias[3 * 64 + cb * 16 + n];

            // i,f,g,o -> state update (selects only; EXEC stays all-ones)
            #pragma unroll
            for (int r = 0; r < 8; ++r) {
                float ig = sigm(acc[0][r] + b0);
                float fg = sigm(acc[1][r] + b1);
                float gg = tanhf(acc[2][r] + b2);
                float og = sigm(acc[3][r] + b3);
                float cn = fg * creg[cb][r] + ig * gg;
                float hn = og * tanhf(cn);
                const bool act = t < llen[r];        // freeze past word end
                creg[cb][r] = act ? cn : creg[cb][r];
                hreg[cb][r] = act ? hn : hreg[cb][r];
            }
        }

        // publish new h for next step's A fragments (cross-lane transpose via LDS)
        #pragma unroll
        for (int cb = 0; cb < 4; ++cb)
            #pragma unroll
            for (int r = 0; r < 8; ++r)
                s_h[wave][(8 * half + r) * HPITCH + cb * 16 + n] = hreg[cb][r];

        __syncthreads();
    }

    // ---- final hidden states: 8 x b128 per lane, coalesced ----
    #pragma unroll
    for (int j = 0; j < 8; ++j) {
        int idx = j * 128 + lane * 4;              // flat [row 0..15][col 0..63]
        int row = idx >> 6, col = idx & 63;
        v4f v = *(const v4f*)&s_h[wave][row * HPITCH + col];
        *(v4f*)&out[(base + row) * HDIM + col] = v;
    }
}

extern "C" void kernel_launch(void* const* d_in, const int* in_sizes, int n_in,
                              void* d_out, int out_size, void* d_ws, size_t ws_size,
                              hipStream_t stream) {
    const float* emb     = (const float*)d_in[0];
    const float* W_ih    = (const float*)d_in[1];
    const float* W_hh    = (const float*)d_in[2];
    const float* b_ih    = (const float*)d_in[3];
    const float* b_hh    = (const float*)d_in[4];
    const int*   chars   = (const int*)d_in[5];
    const int*   lengths = (const int*)d_in[6];
    float* out = (float*)d_out;

    const int n_words = in_sizes[6];               // 65536
    const int blocks  = n_words / (WAVES * 16);    // 128 words per block
    char_lstm_wmma<<<blocks, WAVES * 32, 0, stream>>>(
        emb, W_ih, W_hh, b_ih, b_hh, chars, lengths, out);
}